// CartesianToDihedral_74268574482976
// MI455X (gfx1250) — compile-verified
//
#include <hip/hip_runtime.h>
#include <cstdint>

// ---------------- problem constants (fixed shapes from reference) ----------
#define B_BATCH    1024
#define N_PTS      6144                 // 2048*3 points per batch row
#define N_FLT_ROW  (N_PTS * 3)          // 18432 floats per row
#define N_DIH      (N_PTS - 3)          // 6141 dihedrals per row
#define OUT_ROW    (2 * N_DIH)          // 12282 outputs per row (sin || cos)
#define CHUNKS     4                    // blocks per batch row
#define CHUNK      1536                 // dihedrals per block (last chunk: 1533)
#define THREADS    256
#define MAX_FLT_CHUNK (3 * (CHUNK + 3)) // 4617 floats staged per block

#define EPS_SHIFT_F 1e-8f
#define EPS_NORM_F  1e-12f

#define GLOBAL_AS __attribute__((address_space(1)))
#define LDS_AS    __attribute__((address_space(3)))

typedef int v4i __attribute__((ext_vector_type(4)));

#if defined(__gfx1250__) && \
    __has_builtin(__builtin_amdgcn_global_load_async_to_lds_b128) && \
    __has_builtin(__builtin_amdgcn_global_load_async_to_lds_b32)
#define USE_ASYNC_LDS 1
#else
#define USE_ASYNC_LDS 0
#endif

__global__ __launch_bounds__(THREADS)
void dihedral_kernel(const float* __restrict__ in, float* __restrict__ out)
{
    // 4620 floats = 18480 B of LDS, 16B aligned for B128 async loads
    __shared__ __align__(16) float s_x[((MAX_FLT_CHUNK + 3) / 4) * 4];

    const int bid  = blockIdx.x;
    const int tid  = threadIdx.x;
    const int b    = bid >> 2;                 // batch row
    const int cidx = bid & 3;                  // chunk within row
    const int i0   = cidx * CHUNK;             // first dihedral index of chunk
    const int cnt  = min(CHUNK, N_DIH - i0);   // dihedrals in this chunk
    const int nfl  = 3 * (cnt + 3);            // floats to stage (points i0..i0+cnt+2)

    const float* gsrc = in + (size_t)b * N_FLT_ROW + (size_t)3 * i0; // 16B aligned

    // ---------------- stage chunk into LDS ----------------
#if USE_ASYNC_LDS
    {
        GLOBAL_AS v4i* g4 = (GLOBAL_AS v4i*)gsrc;
        GLOBAL_AS int* gi = (GLOBAL_AS int*)gsrc;
        LDS_AS char*   lb = (LDS_AS char*)(char*)&s_x[0];
        LDS_AS v4i*    l4 = (LDS_AS v4i*)lb;
        LDS_AS int*    li = (LDS_AS int*)lb;

        const int nv4 = nfl >> 2;
        for (int k = tid; k < nv4; k += THREADS)
            __builtin_amdgcn_global_load_async_to_lds_b128(g4 + k, l4 + k, 0, 0);
        const int rem = nfl & 3;  // tail floats (chunk sizes aren't 4-aligned)
        if (tid < rem) {
            const int t = nv4 * 4 + tid;
            __builtin_amdgcn_global_load_async_to_lds_b32(gi + t, li + t, 0, 0);
        }
#if __has_builtin(__builtin_amdgcn_s_wait_asynccnt)
        __builtin_amdgcn_s_wait_asynccnt(0);
#else
        asm volatile("s_wait_asynccnt 0" ::: "memory");
#endif
    }
#else
    {
        const int nv4 = nfl >> 2;
        for (int k = tid; k < nv4; k += THREADS)
            ((float4*)s_x)[k] = ((const float4*)gsrc)[k];
        const int rem = nfl & 3;
        if (tid < rem)
            s_x[nv4 * 4 + tid] = gsrc[nv4 * 4 + tid];
    }
#endif
    __syncthreads();

    // ---------------- compute dihedrals ----------------
    const size_t obase = (size_t)b * OUT_ROW + (size_t)i0;
    float* __restrict__ osin = out + obase;
    float* __restrict__ ocos = out + obase + N_DIH;

    for (int j = tid; j < cnt; j += THREADS) {
        const int p = 3 * j;
        const float ax = s_x[p + 0],  ay = s_x[p + 1],  az = s_x[p + 2];
        const float bx = s_x[p + 3],  by = s_x[p + 4],  bz = s_x[p + 5];
        const float cx = s_x[p + 6],  cy = s_x[p + 7],  cz = s_x[p + 8];
        const float dx = s_x[p + 9],  dy = s_x[p + 10], dz = s_x[p + 11];

        // bc = normalize(b - c + eps)
        const float ux = bx - cx + EPS_SHIFT_F;
        const float uy = by - cy + EPS_SHIFT_F;
        const float uz = bz - cz + EPS_SHIFT_F;
        const float nrm = sqrtf(ux * ux + uy * uy + uz * uz);
        const float inv = 1.0f / fmaxf(nrm, EPS_NORM_F);
        const float tx = ux * inv, ty = uy * inv, tz = uz * inv;

        const float abx = ax - bx, aby = ay - by, abz = az - bz;
        const float cdx = cx - dx, cdy = cy - dy, cdz = cz - dz;

        // n1 = (a-b) x bc
        const float n1x = aby * tz - abz * ty;
        const float n1y = abz * tx - abx * tz;
        const float n1z = abx * ty - aby * tx;
        // n2 = bc x (c-d)
        const float n2x = ty * cdz - tz * cdy;
        const float n2y = tz * cdx - tx * cdz;
        const float n2z = tx * cdy - ty * cdx;

        const float xc = n1x * n2x + n1y * n2y + n1z * n2z;
        // m = n1 x bc
        const float mx = n1y * tz - n1z * ty;
        const float my = n1z * tx - n1x * tz;
        const float mz = n1x * ty - n1y * tx;
        const float yc = mx * n2x + my * n2y + mz * n2z;

        // sin/cos of atan2(yc, xc+eps) without transcendentals:
        const float xs = xc + EPS_SHIFT_F;
        const float rr = xs * xs + yc * yc;
        const float ri = rsqrtf(rr);                      // v_rsq_f32
        const float sv = (rr > 0.0f) ? yc * ri : 0.0f;
        const float cv = (rr > 0.0f) ? xs * ri : 1.0f;    // atan2(0,0)=0

        __builtin_nontemporal_store(sv, osin + j);
        __builtin_nontemporal_store(cv, ocos + j);
    }

    // ---------------- first_three output (one chunk per row owns it) -------
    if (cidx == 0 && tid < 9) {
        float* ft = out + (size_t)B_BATCH * OUT_ROW + (size_t)b * 9;
        ft[tid] = s_x[tid];
    }
}

extern "C" void kernel_launch(void* const* d_in, const int* in_sizes, int n_in,
                              void* d_out, int out_size, void* d_ws, size_t ws_size,
                              hipStream_t stream)
{
    (void)in_sizes; (void)n_in; (void)out_size; (void)d_ws; (void)ws_size;
    const float* in  = (const float*)d_in[0];
    float*       out = (float*)d_out;
    dim3 grid(B_BATCH * CHUNKS);
    dim3 block(THREADS);
    hipLaunchKernelGGL(dihedral_kernel, grid, block, 0, stream, in, out);
}